// SkeletonizationBlock_75600014344430
// MI455X (gfx1250) — compile-verified
//
#include <hip/hip_runtime.h>
#include <hip/hip_bf16.h>
#include <math.h>

typedef __attribute__((ext_vector_type(2))) float v2f;
typedef __attribute__((ext_vector_type(8))) float v8f;
typedef int v4i __attribute__((vector_size(16)));

#define KNN 32
#define PTS_BLK 512
#define CPB 128          // centers per workgroup (8 waves x 16)

// ---- CDNA5 async global->LDS copy (ASYNCcnt-tracked), 16B per lane ----
__device__ __forceinline__ void async_copy_b128(const float* gsrc, float* ldst) {
#if __has_builtin(__builtin_amdgcn_global_load_async_to_lds_b128)
    typedef __attribute__((address_space(1))) v4i* gptr_t;   // global (AS1)
    typedef __attribute__((address_space(3))) v4i* lptr_t;   // LDS (AS3)
    __builtin_amdgcn_global_load_async_to_lds_b128(
        (gptr_t)gsrc, (lptr_t)ldst, 0 /*imm offset*/, 0 /*cpol*/);
#else
    unsigned lds_addr = (unsigned)(unsigned long long)ldst;  // low 32 bits = LDS offset
    asm volatile("global_load_async_to_lds_b128 %0, %1, off"
                 :: "v"(lds_addr), "v"(gsrc) : "memory");
#endif
}

__device__ __forceinline__ void wait_async0() {
#if __has_builtin(__builtin_amdgcn_s_wait_asynccnt)
    __builtin_amdgcn_s_wait_asynccnt(0);
#else
    asm volatile("s_wait_asynccnt 0x0" ::: "memory");
#endif
}

// One workgroup: 128 centers vs all M points.
// WMMA f32 16x16x4: A = points [x,y,z,|x|^2] (16x4), B = centers [-2cx,-2cy,-2cz,1] (4x16)
// D[p,c] = |x_p|^2 - 2 c.x_p  (== d2 minus per-center constant |c|^2; ordering-preserving)
__global__ __launch_bounds__(256)
void skel_knn_kernel(const float* __restrict__ centers,
                     const float* __restrict__ xyz,
                     float* __restrict__ out,
                     int N, int M, int TC)
{
    __shared__ __align__(16) float s_pts[2][PTS_BLK * 3];  // double-buffered staged points
    __shared__ float s_d2[CPB * KNN];      // running top-32 (partial d2)
    __shared__ int   s_id[CPB * KNN];      // neighbor indices
    __shared__ float s_mx[CPB];            // current worst (max) of list
    __shared__ int   s_ag[CPB];            // its position

    const int  t    = threadIdx.x;
    const int  l16  = t & 15;
    const bool lo   = (t & 16) == 0;       // lower half of wave32
    const int  wave = t >> 5;
    const int  slot = (wave << 4) | l16;   // 0..127: which center list this lane serves
    const int  g    = blockIdx.x * CPB + slot;   // global center id
    const int  b    = g / N;
    const float* xyzB = xyz + (size_t)b * M * 3;

    // init top-k state
    for (int i = t; i < CPB * KNN; i += 256) { s_d2[i] = 3.0e38f; s_id[i] = 0; }
    if (t < CPB) { s_mx[t] = 3.0e38f; s_ag[t] = 0; }

    // B operand (4x16, 32-bit): lanes 0-15 hold K=0,1 ; lanes 16-31 hold K=2,3 ; N = lane%16
    const float cx = centers[g * 3 + 0];
    const float cy = centers[g * 3 + 1];
    const float cz = centers[g * 3 + 2];
    v2f Bop;
    Bop.x = lo ? (-2.0f * cx) : (-2.0f * cz);
    Bop.y = lo ? (-2.0f * cy) : 1.0f;

    // prefetch block 0 into buffer 0 (384 x b128 = 6KB)
    for (int i = t; i < (PTS_BLK * 3) / 4; i += 256)
        async_copy_b128(xyzB + i * 4, &s_pts[0][i * 4]);

    float cmx = 3.0e38f;   // register upper bound of s_mx[slot]
    int buf = 0;

    for (int blk = 0; blk < M; blk += PTS_BLK) {
        wait_async0();        // my async writes for this buffer are done
        __syncthreads();      // everyone's writes visible; nobody still reads other buffer

        if (blk + PTS_BLK < M) {
            const float* src = xyzB + (size_t)(blk + PTS_BLK) * 3;
            for (int i = t; i < (PTS_BLK * 3) / 4; i += 256)
                async_copy_b128(src + i * 4, &s_pts[buf ^ 1][i * 4]);
        }

        const float* pts = s_pts[buf];
        for (int ch = 0; ch < PTS_BLK / 16; ++ch) {
            // A operand (16x4, 32-bit): lane L holds row M=L%16; lanes 0-15: K=0,1  lanes 16-31: K=2,3
            const int lp = ch * 16 + l16;
            const float px = pts[lp * 3 + 0];
            const float py = pts[lp * 3 + 1];
            const float pz = pts[lp * 3 + 2];
            const float pw = px * px + py * py + pz * pz;
            v2f Aop;
            Aop.x = lo ? px : pz;
            Aop.y = lo ? py : pw;

            v8f acc = {0.f, 0.f, 0.f, 0.f, 0.f, 0.f, 0.f, 0.f};
            acc = __builtin_amdgcn_wmma_f32_16x16x4_f32(false, Aop, false, Bop,
                                                        (short)0, acc, false, false);

            // D layout: VGPR r -> row M=r (lanes 0-15) or M=8+r (lanes 16-31), col N = lane%16
            const int pbase = blk + ch * 16 + (lo ? 0 : 8);

            // two serialized passes so lanes L and L+16 never touch the same list together
            #pragma unroll
            for (int pass = 0; pass < 2; ++pass) {
                if ((pass == 0) == lo) {
                    #pragma unroll
                    for (int r = 0; r < 8; ++r) {
                        const float d = acc[r];
                        if (d < cmx) {                     // fast path: register compare only
                            const float actual = s_mx[slot];
                            cmx = actual;
                            if (d < actual) {
                                const int a = s_ag[slot];
                                s_d2[slot * KNN + a] = d;
                                s_id[slot * KNN + a] = pbase + r;
                                float mv = -3.0e38f; int am = 0;
                                for (int j = 0; j < KNN; ++j) {
                                    const float v = s_d2[slot * KNN + j];
                                    if (v > mv) { mv = v; am = j; }
                                }
                                s_mx[slot] = mv; s_ag[slot] = am;
                                cmx = mv;
                            }
                        }
                    }
                }
            }
        }
        buf ^= 1;
    }
    __syncthreads();

    // ---- Epilogue: one thread per center ----
    if (t < CPB) {
        const int s  = t;
        const int gc = blockIdx.x * CPB + s;
        const float ccx = centers[gc * 3 + 0];
        const float ccy = centers[gc * 3 + 1];
        const float ccz = centers[gc * 3 + 2];
        const float cc  = ccx * ccx + ccy * ccy + ccz * ccz;
        const int bb = gc / N;
        const float* xb = xyz + (size_t)bb * M * 3;

        const float w1 = 1.0f / (32.0f + 1e-7f);

        // pass 1: weighted mean (uniform weights)
        float mx = 0.f, my = 0.f, mz = 0.f;
        for (int k = 0; k < KNN; ++k) {
            const int id = s_id[s * KNN + k];
            mx += xb[id * 3 + 0];
            my += xb[id * 3 + 1];
            mz += xb[id * 3 + 2];
        }
        mx *= w1; my *= w1; mz *= w1;

        // pass 2: weighted covariance
        float a00 = 0.f, a01 = 0.f, a02 = 0.f, a11 = 0.f, a12 = 0.f, a22 = 0.f;
        for (int k = 0; k < KNN; ++k) {
            const int id = s_id[s * KNN + k];
            const float dx = xb[id * 3 + 0] - mx;
            const float dy = xb[id * 3 + 1] - my;
            const float dz = xb[id * 3 + 2] - mz;
            a00 += w1 * dx * dx; a01 += w1 * dx * dy; a02 += w1 * dx * dz;
            a11 += w1 * dy * dy; a12 += w1 * dy * dz; a22 += w1 * dz * dz;
        }

        // middle eigenvalue of symmetric 3x3 (closed form)
        const float q  = (a00 + a11 + a22) * (1.0f / 3.0f);
        const float p1 = a01 * a01 + a02 * a02 + a12 * a12;
        const float d0 = a00 - q, d1 = a11 - q, d2v = a22 - q;
        const float p2 = d0 * d0 + d1 * d1 + d2v * d2v + 2.0f * p1;
        const float p  = sqrtf(p2 * (1.0f / 6.0f));
        float lmid;
        if (p < 1e-12f) {
            lmid = q;
        } else {
            const float ip  = 1.0f / p;
            const float b00 = d0 * ip, b11 = d1 * ip, b22 = d2v * ip;
            const float b01 = a01 * ip, b02 = a02 * ip, b12 = a12 * ip;
            const float detB = b00 * (b11 * b22 - b12 * b12)
                             - b01 * (b01 * b22 - b12 * b02)
                             + b02 * (b01 * b12 - b11 * b02);
            float r = 0.5f * detB;
            r = fminf(1.0f, fmaxf(-1.0f, r));
            const float phi = acosf(r) * (1.0f / 3.0f);
            const float e1 = q + 2.0f * p * __cosf(phi);
            const float e3 = q + 2.0f * p * __cosf(phi + 2.0943951023931953f); // +2pi/3
            lmid = 3.0f * q - e1 - e3;
        }
        const float radius = sqrtf(fmaxf(lmid, 0.0f));
        const float thr = 3.0f * radius;

        // pass 3: re-mask by 3*radius and recompute mean
        float sm = 0.f, ox = 0.f, oy = 0.f, oz = 0.f;
        for (int k = 0; k < KNN; ++k) {
            const int id = s_id[s * KNN + k];
            const float dist = sqrtf(fmaxf(s_d2[s * KNN + k] + cc, 0.0f));
            const float m = (dist <= thr) ? 1.0f : 0.0f;
            sm += m;
            ox += m * xb[id * 3 + 0];
            oy += m * xb[id * 3 + 1];
            oz += m * xb[id * 3 + 2];
        }
        const float inv = 1.0f / (sm + 1e-7f);
        out[gc * 3 + 0] = ox * inv;
        out[gc * 3 + 1] = oy * inv;
        out[gc * 3 + 2] = oz * inv;
        out[(size_t)TC * 3 + gc] = radius;   // radius block after new_centers block
    }
}

extern "C" void kernel_launch(void* const* d_in, const int* in_sizes, int n_in,
                              void* d_out, int out_size, void* d_ws, size_t ws_size,
                              hipStream_t stream) {
    const float* centers = (const float*)d_in[0];   // [B,N,3]
    const float* xyz     = (const float*)d_in[1];   // [B,M,3]
    float* out = (float*)d_out;                     // [B*N*3] new_centers ++ [B*N] radius

    const int B  = 4;
    const int N  = in_sizes[0] / (B * 3);           // 4096
    const int M  = in_sizes[1] / (B * 3);           // 16384
    const int TC = B * N;                           // 16384 centers
    const int nwg = TC / CPB;                       // 128 workgroups

    hipLaunchKernelGGL(skel_knn_kernel, dim3(nwg), dim3(256), 0, stream,
                       centers, xyz, out, N, M, TC);
}